// RNN_seplay_86148454023839
// MI455X (gfx1250) — compile-verified
//
#include <hip/hip_runtime.h>
#include <math.h>

// Problem constants (from reference)
#define BB    32
#define TT    1000
#define NN    512
#define INN   16
#define OUTN  8
#define TN    (TT * NN)     // 512000
#define TIN   (TT * INN)    // 16000
#define ALPHA_C   0.1f
#define NSCALE_C  0.022360679774997896f   // sqrt(2*0.1*0.05^2)
#define FLTMAX_C  3.4028234e38f

// 16 blocks (one per N-tile pair) * 4 waves (side x mtile) * 2 tiles = 128 tiles
#define NBLK      16
#define NTHR      128

// LDS weight cache: 4 sections (side,ntile) x 260 k-steps x 64 floats (256B)
#define KSTEPS    260               // 128 (W main) + 128 (W cross) + 4 (W in)
#define SECF      (KSTEPS * 64)     // floats per section
#define LDS_BYTES (4 * SECF * 4)    // 266240 B = 260 KB (fits 320 KB WGP LDS)

typedef __attribute__((ext_vector_type(2))) float v2f;
typedef __attribute__((ext_vector_type(8))) float v8f;

// Exact-fp32 CDNA5 WMMA: D(16x16,f32) = A(16x4,f32) x B(4x16,f32) + C
__device__ __forceinline__ v8f wmma4(v2f a, v2f b, v8f c) {
  return __builtin_amdgcn_wmma_f32_16x16x4_f32(false, a, false, b, (short)0, c,
                                               false, false);
}

__device__ __forceinline__ v2f ldv2(const float* __restrict__ p, unsigned off) {
  return *(const v2f*)(p + off);
}
__device__ __forceinline__ float ld1(const float* __restrict__ p, unsigned off) {
  return p[off];
}
__device__ __forceinline__ void st1(float* __restrict__ p, unsigned off, float v) {
  p[off] = v;
}
// LDS fragment access (generic pointer derived from shared -> ds ops)
__device__ __forceinline__ v2f ldsv2(const float* p, int off) {
  return *(const v2f*)(p + off);
}
// relu in one VALU op (v_med3), -relu with folded neg modifier
__device__ __forceinline__ v2f relu2(v2f a) {
  v2f r;
  r.x = __builtin_amdgcn_fmed3f(a.x, 0.f, FLTMAX_C);
  r.y = __builtin_amdgcn_fmed3f(a.y, 0.f, FLTMAX_C);
  return r;
}
__device__ __forceinline__ v2f nrelu2(v2f a) {
  v2f r;
  r.x = __builtin_amdgcn_fmed3f(-a.x, -FLTMAX_C, 0.f);
  r.y = __builtin_amdgcn_fmed3f(-a.y, -FLTMAX_C, 0.f);
  return r;
}

// Grid-wide sense-reversal barrier. cnt at bar[0], sense at bar[16].
__device__ __forceinline__ void grid_barrier(unsigned* bar, unsigned* myphase) {
  __builtin_amdgcn_fence(__ATOMIC_RELEASE, "agent");   // flush WGP$ dirty -> L2
  __syncthreads();
  if (threadIdx.x == 0) {
    unsigned target = *myphase + 1u;
    unsigned old = __hip_atomic_fetch_add(&bar[0], 1u, __ATOMIC_ACQ_REL,
                                          __HIP_MEMORY_SCOPE_AGENT);
    if (old == NBLK - 1u) {
      __hip_atomic_store(&bar[0], 0u, __ATOMIC_RELAXED, __HIP_MEMORY_SCOPE_AGENT);
      __hip_atomic_store(&bar[16], target, __ATOMIC_RELEASE, __HIP_MEMORY_SCOPE_AGENT);
    } else {
      while (__hip_atomic_load(&bar[16], __ATOMIC_ACQUIRE,
                               __HIP_MEMORY_SCOPE_AGENT) < target) {
        __builtin_amdgcn_s_sleep(1);
      }
    }
    *myphase = target;
  }
  __syncthreads();
  __builtin_amdgcn_fence(__ATOMIC_ACQUIRE, "agent");   // invalidate stale lines
}

extern __shared__ float ldsbuf[];

__global__ void __launch_bounds__(NTHR)
rnn_recurrence_kernel(const float* __restrict__ u,
                      const float* __restrict__ in0,
                      const float* __restrict__ e0,
                      const float* __restrict__ i0,
                      const float* __restrict__ pE, const float* __restrict__ pI,
                      const float* __restrict__ nE, const float* __restrict__ nI,
                      const float* __restrict__ Wee, const float* __restrict__ Wei,
                      const float* __restrict__ Wie, const float* __restrict__ Wii,
                      const float* __restrict__ Wine, const float* __restrict__ Wini,
                      float* __restrict__ Sin,
                      float* __restrict__ AE, float* __restrict__ AI,
                      float* __restrict__ SE, float* __restrict__ SI,
                      unsigned* __restrict__ bar) {
  const int gtid = blockIdx.x * NTHR + threadIdx.x;      // 0..2047
  unsigned myphase = 0;

  // ---- Init phase: t=0 states + relu acts, and the full states_in scan ----
  for (int idx = gtid; idx < 2 * BB * NN; idx += NBLK * NTHR) {
    int which = idx / (BB * NN);
    int r = idx % (BB * NN);
    float v = which ? i0[r] : e0[r];
    float* S = which ? SI : SE;
    float* A = which ? AI : AE;
    int b = r / NN, n = r % NN;
    st1(S, (unsigned)(b * TN + n), v);
    st1(A, (unsigned)(b * TN + n), fmaxf(v, 0.f));
  }
  if (gtid < BB * INN) {                  // 512 tiny independent scans
    int b = gtid / INN, j = gtid % INN;
    float s = in0[gtid];
    unsigned ub = (unsigned)(b * TIN + j);
    st1(Sin, ub, s);
#pragma unroll 4
    for (int t = 0; t < TT - 1; ++t) {
      s = (1.f - ALPHA_C) * s + ALPHA_C * ld1(u, ub + (unsigned)(t * INN));
      st1(Sin, ub + (unsigned)((t + 1) * INN), s);
    }
  }

  const int lane = threadIdx.x & 31;
  const int hi   = lane >> 4;
  const int lm   = lane & 15;
  const int w    = threadIdx.x >> 5;      // wave in block: 0..3
  const int l8   = lane * 2;              // float offset of this lane's LDS slot

  // ---- Fill LDS weight cache (once). Wave w fills section w = (side, p). ----
  {
    const int fs = w >> 1;                // section side
    const int fp = w & 1;                 // section ntile within pair
    const int nt = blockIdx.x * 2 + fp;
    const float* Wa_s = fs ? Wei : Wee;
    const float* Wb_s = fs ? Wii : Wie;
    const float* Wc_s = fs ? Wini : Wine;
    const unsigned gw = (unsigned)((nt * 16 + lm) * NN + 2 * hi);
    const unsigned gc = (unsigned)((nt * 16 + lm) * INN + 2 * hi);
    float* sec = ldsbuf + w * SECF;
#pragma unroll 4
    for (int j = 0; j < 128; ++j)
      *(v2f*)(sec + j * 64 + l8) = ldv2(Wa_s, gw + 4 * j);
#pragma unroll 4
    for (int j = 0; j < 128; ++j)
      *(v2f*)(sec + (128 + j) * 64 + l8) = ldv2(Wb_s, gw + 4 * j);
#pragma unroll
    for (int j = 0; j < 4; ++j)
      *(v2f*)(sec + (256 + j) * 64 + l8) = ldv2(Wc_s, gc + 4 * j);
  }

  // ---- Compute-role assignment: wave w -> (side, mtile); 2 N-tiles/wave ----
  const int side  = w >> 1;
  const int mtile = w & 1;
  const int nt0   = blockIdx.x * 2;
  const unsigned bM = (unsigned)(mtile * 16 + lm);   // A-fragment batch row

  const float* Pp = side ? pI : pE;
  const float* Zp = side ? nI : nE;
  float* Sd = side ? SI : SE;             // new state dst (== old state src)
  float* Ad = side ? AI : AE;

  // LDS section bases for this side's two N-tiles, split per K segment
  const float* sec0 = ldsbuf + (side * 2 + 0) * SECF;
  const float* sec1 = ldsbuf + (side * 2 + 1) * SECF;
  const float* sA0 = sec0;              const float* sA1 = sec1;
  const float* sB0 = sec0 + 128 * 64;   const float* sB1 = sec1 + 128 * 64;
  const float* sC0 = sec0 + 256 * 64;   const float* sC1 = sec1 + 256 * 64;

  // Epilogue row/col offsets (C layout): VGPR r -> row mtile*16 + r + 8*hi
  unsigned rowOff[8];
#pragma unroll
  for (int r = 0; r < 8; ++r)
    rowOff[r] = (unsigned)((mtile * 16 + r + hi * 8) * TN + nt0 * 16 + lm);

  grid_barrier(bar, &myphase);            // t=0 state + LDS fill visible

  // Leak-term state carried in registers (we wrote/own this tile location).
  float old0[8], old1[8];
#pragma unroll
  for (int r = 0; r < 8; ++r) {
    old0[r] = ld1(Sd, rowOff[r]);
    old1[r] = ld1(Sd, rowOff[r] + 16);
  }

  const unsigned aE0 = bM * (unsigned)TN + (unsigned)(2 * hi);
  const unsigned aN0 = bM * (unsigned)TIN + (unsigned)(2 * hi);

  // ---- Main sequential loop ----
  for (int t = 0; t < TT - 1; ++t) {
    const unsigned tn  = (unsigned)t * NN;
    const unsigned aOe = aE0 + tn;                 // into SE / SI
    const unsigned aOn = aN0 + (unsigned)t * INN;  // into Sin

    // Prefetch perturb/noise so they overlap the matmul.
    float pv[16], zv[16];
#pragma unroll
    for (int r = 0; r < 8; ++r) {
      pv[r]     = ld1(Pp, rowOff[r] + tn);
      pv[r + 8] = ld1(Pp, rowOff[r] + tn + 16);
      zv[r]     = ld1(Zp, rowOff[r] + tn);
      zv[r + 8] = ld1(Zp, rowOff[r] + tn + 16);
    }

    v8f acc0 = {0.f, 0.f, 0.f, 0.f, 0.f, 0.f, 0.f, 0.f};
    v8f acc1 = {0.f, 0.f, 0.f, 0.f, 0.f, 0.f, 0.f, 0.f};

    // Segment 1: reluE @ Wa^T   (A global pipelined; B from LDS)
    {
      v2f aC = ldv2(SE, aOe);
#pragma unroll 4
      for (int j = 0; j < 127; ++j) {
        v2f aN = ldv2(SE, aOe + 4 * (j + 1));
        v2f ar = relu2(aC);
        acc0 = wmma4(ar, ldsv2(sA0, j * 64 + l8), acc0);
        acc1 = wmma4(ar, ldsv2(sA1, j * 64 + l8), acc1);
        aC = aN;
      }
      v2f ar = relu2(aC);
      acc0 = wmma4(ar, ldsv2(sA0, 127 * 64 + l8), acc0);
      acc1 = wmma4(ar, ldsv2(sA1, 127 * 64 + l8), acc1);
    }
    // Segment 2: (-reluI) @ Wb^T
    {
      v2f aC = ldv2(SI, aOe);
#pragma unroll 4
      for (int j = 0; j < 127; ++j) {
        v2f aN = ldv2(SI, aOe + 4 * (j + 1));
        v2f ar = nrelu2(aC);
        acc0 = wmma4(ar, ldsv2(sB0, j * 64 + l8), acc0);
        acc1 = wmma4(ar, ldsv2(sB1, j * 64 + l8), acc1);
        aC = aN;
      }
      v2f ar = nrelu2(aC);
      acc0 = wmma4(ar, ldsv2(sB0, 127 * 64 + l8), acc0);
      acc1 = wmma4(ar, ldsv2(sB1, 127 * 64 + l8), acc1);
    }
    // Segment 3: s_in @ Wc^T  (K = 16)
#pragma unroll
    for (int j = 0; j < 4; ++j) {
      v2f a = ldv2(Sin, aOn + 4 * j);
      acc0 = wmma4(a, ldsv2(sC0, j * 64 + l8), acc0);
      acc1 = wmma4(a, ldsv2(sC1, j * 64 + l8), acc1);
    }

    // Epilogue: leak + perturb + noise; write state(t+1) and relu act(t+1).
#pragma unroll
    for (int r = 0; r < 8; ++r) {
      const unsigned d0 = rowOff[r] + tn + NN;         // (t+1) slot, col c0
      const unsigned d1 = d0 + 16;                     // col c1
      float nv0 = (1.f - ALPHA_C) * old0[r] + ALPHA_C * (acc0[r] + pv[r])
                  + NSCALE_C * zv[r];
      float nv1 = (1.f - ALPHA_C) * old1[r] + ALPHA_C * (acc1[r] + pv[r + 8])
                  + NSCALE_C * zv[r + 8];
      st1(Sd, d0, nv0);
      st1(Ad, d0, fmaxf(nv0, 0.f));
      st1(Sd, d1, nv1);
      st1(Ad, d1, fmaxf(nv1, 0.f));
      old0[r] = nv0;
      old1[r] = nv1;
    }
    grid_barrier(bar, &myphase);
  }
}

// out[b,t,:] = relu_e[b,t,:] @ W_out^T over all 32000 flattened rows.
// N padded 8->16 via zeroed ws region for dead B lanes; dead C columns dropped.
__global__ void __launch_bounds__(128)
rnn_out_kernel(const float* __restrict__ acte, const float* __restrict__ wout,
               const float* __restrict__ zpad, float* __restrict__ out) {
  const int lane = threadIdx.x & 31;
  const int hi = lane >> 4, lm = lane & 15;
  const int gw = blockIdx.x * 4 + (threadIdx.x >> 5);   // 0..1999
  const int m0 = gw * 16;                               // (b*T+t) row tile base
  const float* Ab = acte + (size_t)(m0 + lm) * NN + 2 * hi;
  const float* Bb = (lm < OUTN) ? (wout + (size_t)lm * NN + 2 * hi)
                                : (zpad + 2 * hi);
  v8f acc = {0.f, 0.f, 0.f, 0.f, 0.f, 0.f, 0.f, 0.f};
  v2f aC = *(const v2f*)(Ab);
  v2f bC = *(const v2f*)(Bb);
#pragma unroll 4
  for (int k = 4; k < NN; k += 4) {
    v2f aN = *(const v2f*)(Ab + k);
    v2f bN = *(const v2f*)(Bb + k);
    acc = wmma4(aC, bC, acc);
    aC = aN; bC = bN;
  }
  acc = wmma4(aC, bC, acc);
  if (lm < OUTN) {
#pragma unroll
    for (int r = 0; r < 8; ++r) {
      const int row = m0 + r + hi * 8;
      out[(size_t)row * OUTN + lm] = acc[r];
    }
  }
}

extern "C" void kernel_launch(void* const* d_in, const int* in_sizes, int n_in,
                              void* d_out, int out_size, void* d_ws, size_t ws_size,
                              hipStream_t stream) {
  (void)in_sizes; (void)n_in; (void)out_size; (void)ws_size;
  const float* u    = (const float*)d_in[0];
  const float* in0  = (const float*)d_in[1];
  const float* e0   = (const float*)d_in[2];
  const float* i0   = (const float*)d_in[3];
  const float* pE   = (const float*)d_in[4];
  const float* pI   = (const float*)d_in[5];
  const float* nE   = (const float*)d_in[6];
  const float* nI   = (const float*)d_in[7];
  const float* Wee  = (const float*)d_in[8];
  const float* Wei  = (const float*)d_in[9];
  const float* Wie  = (const float*)d_in[10];
  const float* Wii  = (const float*)d_in[11];
  const float* Wine = (const float*)d_in[12];
  const float* Wini = (const float*)d_in[13];
  const float* Wout = (const float*)d_in[14];

  // d_out regions (return order): out, states_in, act_e, act_i, states_e, states_i
  float* out = (float*)d_out;
  float* Sin = out + (size_t)BB * TT * OUTN;
  float* AE  = Sin + (size_t)BB * TT * INN;
  float* AI  = AE + (size_t)BB * TT * NN;
  float* SE  = AI + (size_t)BB * TT * NN;
  float* SI  = SE + (size_t)BB * TT * NN;

  // ws: [0..63] barrier cnt, [64..127] sense, [1024..] 512-float zero pad.
  (void)hipMemsetAsync(d_ws, 0, 8192, stream);          // graph-capture safe
  unsigned* bar = (unsigned*)d_ws;
  const float* zpad = (const float*)((char*)d_ws + 1024);

  // Allow the 260KB dynamic-LDS allocation (no-op if already permitted).
  static bool attr_done = false;
  if (!attr_done) {
    (void)hipFuncSetAttribute((const void*)rnn_recurrence_kernel,
                              hipFuncAttributeMaxDynamicSharedMemorySize,
                              LDS_BYTES);
    attr_done = true;
  }

  rnn_recurrence_kernel<<<NBLK, NTHR, LDS_BYTES, stream>>>(
      u, in0, e0, i0, pE, pI, nE, nI, Wee, Wei, Wie, Wii, Wine, Wini,
      Sin, AE, AI, SE, SI, bar);
  rnn_out_kernel<<<500, 128, 0, stream>>>(AE, Wout, zpad, out);
}